// GCNConv_68143951118618
// MI455X (gfx1250) — compile-verified
//
#include <hip/hip_runtime.h>
#include <stdint.h>

typedef float v2f __attribute__((ext_vector_type(2)));
typedef float v8f __attribute__((ext_vector_type(8)));

#define DIM 64          // D_IN == D_OUT == 64
#define LDSW (DIM + 1)  // padded stride to avoid LDS bank conflicts

// ---------------------------------------------------------------------------
// Stage 1: h = x @ W^T + b   (fp32 WMMA 16x16x4)
//   One wave -> one 16-row node tile, all 4 column tiles (64 outputs).
//   A(16x4):  lane m = lane&15, K = kk + 2*(lane>>4) + {0,1}
//   B(4x16):  lane n = lane&15, same K split  (B[k][n] = W[n][k])
//   C/D:      N = lane&15, M = r + 8*(lane>>4), r = vgpr 0..7
// ---------------------------------------------------------------------------
__global__ __launch_bounds__(256) void gcn_gemm(const float* __restrict__ x,
                                                const float* __restrict__ W,
                                                const float* __restrict__ bias,
                                                float* __restrict__ h,
                                                int n_nodes)
{
    __shared__ float sW[DIM * LDSW];   // 16.25 KB of the 320 KB LDS

    const int tid = threadIdx.x;
    for (int i = tid; i < DIM * DIM; i += 256) {
        int o = i >> 6;
        int k = i & 63;
        sW[o * LDSW + k] = W[i];
    }
    __syncthreads();

    const int wave = tid >> 5;
    const int lane = tid & 31;
    const long rowbase = ((long)blockIdx.x * 8 + wave) * 16;
    if (rowbase >= n_nodes) return;    // wave-uniform; full-EXEC for WMMA

    const int m     = lane & 15;       // row (A) / col (B,D) index within tile
    const int khalf = lane >> 4;       // which K pair this lane holds

    v8f acc[4] = {};                   // 4 x (16x16) fp32 output tiles

    const float* __restrict__ xrow = x + (rowbase + m) * DIM;

#pragma unroll
    for (int kk = 0; kk < DIM; kk += 4) {
        const int k0 = kk + 2 * khalf;
        const v2f a = *(const v2f*)(xrow + k0);     // A fragment (2 consecutive K)
#pragma unroll
        for (int t = 0; t < 4; ++t) {
            const int o = t * 16 + m;               // output feature column
            v2f bf;
            bf.x = sW[o * LDSW + k0];
            bf.y = sW[o * LDSW + k0 + 1];
            acc[t] = __builtin_amdgcn_wmma_f32_16x16x4_f32(
                false, a, false, bf, (short)0, acc[t], false, false);
        }
    }

    // Write h tile + bias.  Row for vgpr r is rowbase + r + 8*khalf.
#pragma unroll
    for (int t = 0; t < 4; ++t) {
        const int o = t * 16 + m;
        const float bv = bias[o];
#pragma unroll
        for (int r = 0; r < 8; ++r) {
            const long row = rowbase + r + 8 * khalf;
            h[row * DIM + o] = acc[t][r] + bv;
        }
    }
}

// ---------------------------------------------------------------------------
// Stage 2: zero the output accumulator (harness poisons d_out each run)
// ---------------------------------------------------------------------------
__global__ __launch_bounds__(256) void zero_out(float4* __restrict__ p, long n4)
{
    const long i = (long)blockIdx.x * blockDim.x + threadIdx.x;
    if (i < n4) p[i] = make_float4(0.f, 0.f, 0.f, 0.f);
}

// ---------------------------------------------------------------------------
// Stage 3: out[dst] += w_e * h[src]   (16 lanes per edge, float4 per lane)
//   256 B coalesced gather of h[src]; 4 global_atomic_add_f32 per lane.
//   out (25.6 MB) is L2-resident (192 MB), so atomics stay on-chip.
// ---------------------------------------------------------------------------
__global__ __launch_bounds__(256) void gcn_scatter(const long long* __restrict__ ei,
                                                   const float* __restrict__ ew,
                                                   const float* __restrict__ h,
                                                   float* __restrict__ out,
                                                   int n_edges)
{
    const long gt = (long)blockIdx.x * blockDim.x + threadIdx.x;
    const long e  = gt >> 4;            // edge id (16 lanes/edge)
    const int  l  = (int)(gt & 15);     // lane within edge group
    if (e >= n_edges) return;

    const long long dst = ei[e];                    // edge_index[0][e] (row)
    const long long src = ei[(long)n_edges + e];    // edge_index[1][e] (col)
    const float     w   = ew[e];

    const float4 v = ((const float4*)(h + (long)src * DIM))[l];
    float* o = out + (long)dst * DIM + l * 4;
    atomicAdd(o + 0, v.x * w);
    atomicAdd(o + 1, v.y * w);
    atomicAdd(o + 2, v.z * w);
    atomicAdd(o + 3, v.w * w);
}

// ---------------------------------------------------------------------------
extern "C" void kernel_launch(void* const* d_in, const int* in_sizes, int n_in,
                              void* d_out, int out_size, void* d_ws, size_t ws_size,
                              hipStream_t stream)
{
    // setup_inputs order: x, edge_index(int64), edge_weight, num_nodes, W, b
    const float*     x  = (const float*)d_in[0];
    const long long* ei = (const long long*)d_in[1];
    const float*     ew = (const float*)d_in[2];
    const float*     W  = (const float*)d_in[4];
    const float*     b  = (const float*)d_in[5];

    const int n_nodes = in_sizes[0] / DIM;   // 100000
    const int n_edges = in_sizes[2];         // 1.6M

    float* h   = (float*)d_ws;               // [n_nodes, 64] fp32 = 25.6 MB
    float* out = (float*)d_out;

    // Stage 2 first in stream order (independent of GEMM).
    const long n4 = (long)out_size / 4;
    zero_out<<<(int)((n4 + 255) / 256), 256, 0, stream>>>((float4*)out, n4);

    // Stage 1: 8 waves/block, one 16-row tile per wave.
    const int ntiles = (n_nodes + 15) / 16;
    const int nblk   = (ntiles + 7) / 8;
    gcn_gemm<<<nblk, 256, 0, stream>>>(x, W, b, h, n_nodes);

    // Stage 3: 16 threads per edge.
    const long nthreads = (long)n_edges * 16;
    gcn_scatter<<<(int)((nthreads + 255) / 256), 256, 0, stream>>>(ei, ew, h, out, n_edges);
}